// Max_iou_assigner_9174050144499
// MI455X (gfx1250) — compile-verified
//
#include <hip/hip_runtime.h>
#include <hip/hip_bf16.h>

#define MAX_G 128
#define POS_THR 0.5f
#define NEG_THR 0.4f
#define EPS_F 1e-16f

typedef unsigned int u32;
typedef unsigned long long u64;
typedef __attribute__((ext_vector_type(4))) u32 u32x4_t;
typedef __attribute__((ext_vector_type(8))) int i32x8_t;
typedef __attribute__((ext_vector_type(4))) int i32x4_t;

// ---------------------------------------------------------------------------
// CDNA5 Tensor Data Mover: DMA a 1-D run of `nelem` f32 elements from global
// memory into LDS at byte offset `lds_byte_off`. Descriptor per
// CDNA5 ISA (Group0/Group1; groups 2/3 zero => <=2D).
// ---------------------------------------------------------------------------
__device__ __forceinline__ void tdm_wait0() {
#if __has_builtin(__builtin_amdgcn_s_wait_tensorcnt)
  __builtin_amdgcn_s_wait_tensorcnt(0);
#else
  asm volatile("s_wait_tensorcnt 0x0" ::: "memory");
#endif
}

#if __has_builtin(__builtin_amdgcn_tensor_load_to_lds)
#define HAVE_TDM 1
__device__ __forceinline__ void tdm_load_1d_f32(u32 lds_byte_off, const void* gptr, u32 nelem) {
  u64 ga = (u64)gptr;
  u32x4_t g0;
  g0[0] = 1u;                                             // count=1 (valid user D#)
  g0[1] = lds_byte_off;                                   // lds_addr [63:32]
  g0[2] = (u32)(ga & 0xFFFFFFFFull);                      // global_addr[31:0]
  g0[3] = (u32)((ga >> 32) & 0x1FFFFFFull) | 0x80000000u; // global_addr[56:32] | type=2
  i32x8_t g1;
  g1[0] = (int)(2u << 16);                                // data_size=2 (4 bytes); wg_mask=0
  g1[1] = (int)((nelem & 0xFFFFu) << 16);                 // tensor_dim0[15:0] @ bits 63:48
  g1[2] = (int)(((nelem >> 16) & 0xFFFFu) | (1u << 16));  // dim0[31:16] | tensor_dim1=1
  g1[3] = (int)((nelem & 0xFFFFu) << 16);                 // tile_dim0 @ bits 127:112
  g1[4] = 1;                                              // tile_dim1=1, tile_dim2=0
  g1[5] = (int)nelem;                                     // tensor_dim0_stride[31:0]
  g1[6] = (int)((nelem & 0xFFFFu) << 16);                 // d0s[47:32]=0 | dim1_stride[15:0]
  g1[7] = (int)(nelem >> 16);                             // dim1_stride[47:16]
  i32x4_t z4 = {0, 0, 0, 0};
#if defined(__clang_major__) && __clang_major__ >= 23
  i32x8_t z8 = {0, 0, 0, 0, 0, 0, 0, 0};
  __builtin_amdgcn_tensor_load_to_lds(g0, g1, z4, z4, z8, 0);
#else
  __builtin_amdgcn_tensor_load_to_lds(g0, g1, z4, z4, 0);
#endif
}
#else
#define HAVE_TDM 0
#endif

// Stage n floats global->LDS. TDM path issues one DMA from wave 0.
__device__ __forceinline__ void stage_lds(float* dst, const float* __restrict__ src, int n) {
#if HAVE_TDM
  if ((threadIdx.x >> 5) == 0) {
    tdm_load_1d_f32((u32)(u64)(void*)dst, (const void*)src, (u32)n);
  }
#else
  for (int k = (int)threadIdx.x; k < n; k += (int)blockDim.x) dst[k] = src[k];
#endif
}

__device__ __forceinline__ void stage_fence() {
#if HAVE_TDM
  if ((threadIdx.x >> 5) == 0) tdm_wait0();
#endif
  __syncthreads();
}

// ---------------------------------------------------------------------------
// IoU core with both areas hoisted. The exact same expression tree is used in
// both passes (min/max/sub/mul + IEEE div, no FMA-contraction opportunity),
// and area_b/area_g are produced by the same sub/mul ops on the same inputs
// in both passes, so the pass-2 equality test against pass-1 maxima is
// bit-exact.
// ---------------------------------------------------------------------------
__device__ __forceinline__ float box_area(float x1, float y1, float x2, float y2) {
  return (x2 - x1) * (y2 - y1);
}

__device__ __forceinline__ float iou_core(float bx1, float by1, float bx2, float by2,
                                          float area_b,
                                          float gx1, float gy1, float gx2, float gy2,
                                          float area_g) {
  float iw = fmaxf(fminf(bx2, gx2) - fmaxf(bx1, gx1), 0.0f);
  float ih = fmaxf(fminf(by2, gy2) - fmaxf(by1, gy1), 0.0f);
  float inter = iw * ih;
  return inter / (area_b + area_g - inter + EPS_F);
}

// ---------------------------------------------------------------------------
// Kernel 0: init per-GT column max (int bit-pattern of +0.0f; IoU >= 0 so
// integer atomicMax ordering == float ordering).
// ---------------------------------------------------------------------------
__global__ void init_colmax_kernel(int* __restrict__ colmax, int G) {
  int t = (int)(blockIdx.x * blockDim.x + threadIdx.x);
  if (t < G) colmax[t] = 0;
}

// ---------------------------------------------------------------------------
// Kernel 1: column max. Thread t owns GT column t; blocks grid-stride over
// 128-box tiles staged in LDS (box + precomputed area). Boxes are read
// exactly once from HBM.
// ---------------------------------------------------------------------------
struct P1Shared {
  float gt[5 * MAX_G];
  float4 tile[128];
  float tarea[128];
};

__global__ void __launch_bounds__(128) iou_colmax_kernel(
    const float4* __restrict__ bboxes, const float* __restrict__ targets,
    int* __restrict__ colmax, int N, int G) {
  __shared__ P1Shared sm;
  stage_lds(sm.gt, targets, 5 * G);
  stage_fence();

  int t = (int)threadIdx.x;
  float gx1 = 0.f, gy1 = 0.f, gx2 = 0.f, gy2 = 0.f, area_g = 0.f;
  bool valid = false;
  if (t < G) {
    gx1 = sm.gt[5 * t + 0];
    gy1 = sm.gt[5 * t + 1];
    gx2 = sm.gt[5 * t + 2];
    gy2 = sm.gt[5 * t + 3];
    valid = (sm.gt[5 * t + 4] != -1.0f);
    area_g = box_area(gx1, gy1, gx2, gy2);
  }

  float localmax = 0.0f;
  int nTiles = (N + 127) >> 7;
  for (int tile = (int)blockIdx.x; tile < nTiles; tile += (int)gridDim.x) {
    int base = tile << 7;
    int idx = base + t;
    if (idx < N) {
      float4 b = bboxes[idx];
      sm.tile[t] = b;
      sm.tarea[t] = box_area(b.x, b.y, b.z, b.w);
    }
    // prefetch next tile for this block while we crunch the current one
    int nidx = base + (int)gridDim.x * 128 + t;
    if (nidx < N) __builtin_prefetch((const void*)&bboxes[nidx], 0, 1);
    __syncthreads();
    int cnt = N - base;
    if (cnt > 128) cnt = 128;
    if (valid) {
      for (int i = 0; i < cnt; ++i) {
        float4 b = sm.tile[i];  // LDS broadcast read (conflict-free)
        float v = iou_core(b.x, b.y, b.z, b.w, sm.tarea[i], gx1, gy1, gx2, gy2, area_g);
        localmax = fmaxf(localmax, v);
      }
    }
    __syncthreads();
  }
  if (t < G && valid) atomicMax(&colmax[t], __float_as_int(localmax));
}

// ---------------------------------------------------------------------------
// Kernel 2: per-box assignment. One thread per box. GT table + column maxes
// DMA'd into LDS via TDM, then repacked into two float4 tables so the hot
// loop is two broadcast ds_load_b128 per GT.
// ---------------------------------------------------------------------------
struct P2Shared {
  float raw[5 * MAX_G];
  float craw[MAX_G];
  float4 q0[MAX_G];  // x1, y1, x2, y2
  float4 q1[MAX_G];  // area_g, label (-1 => invalid), colmax, 0
};

__global__ void __launch_bounds__(256) assign_kernel(
    const float4* __restrict__ bboxes, const float* __restrict__ targets,
    const int* __restrict__ colmax,
    float* __restrict__ out_assigned, float* __restrict__ out_max,
    float* __restrict__ out_labels, int N, int G) {
  __shared__ P2Shared sm;
  stage_lds(sm.raw, targets, 5 * G);
  stage_lds(sm.craw, (const float*)colmax, G);
  stage_fence();

  // Pack GT table: {box} and {area_g, label, colmax, 0}
  for (int j = (int)threadIdx.x; j < G; j += (int)blockDim.x) {
    float x1 = sm.raw[5 * j + 0];
    float y1 = sm.raw[5 * j + 1];
    float x2 = sm.raw[5 * j + 2];
    float y2 = sm.raw[5 * j + 3];
    float lb = sm.raw[5 * j + 4];
    sm.q0[j] = make_float4(x1, y1, x2, y2);
    sm.q1[j] = make_float4(box_area(x1, y1, x2, y2), lb, sm.craw[j], 0.0f);
  }
  __syncthreads();

  int i = (int)(blockIdx.x * blockDim.x + threadIdx.x);
  if (i >= N) return;
  float4 b = bboxes[i];
  float area_b = box_area(b.x, b.y, b.z, b.w);

  float rowmax = -3.0e38f;
  int argmax = 0;
  int lastj = -1;
#pragma unroll 4
  for (int j = 0; j < G; ++j) {
    float4 q0 = sm.q0[j];  // broadcast reads
    float4 q1 = sm.q1[j];
    bool valid = (q1.y != -1.0f);
    float v = iou_core(b.x, b.y, b.z, b.w, area_b, q0.x, q0.y, q0.z, q0.w, q1.x);
    float masked = valid ? v : -1.0f;
    if (masked > rowmax) { rowmax = masked; argmax = j; }  // first-max tie-break
    if (valid && (v == q1.z)) lastj = j;                   // last-max tie-break
  }

  int assigned = (rowmax > POS_THR) ? (argmax + 1) : -1;
  if (rowmax < NEG_THR) assigned = 0;
  if (lastj >= 0) assigned = lastj + 1;

  float label = -1.0f;
  if (assigned > 0) label = sm.q1[assigned - 1].y;

  out_assigned[i] = (float)assigned;
  out_max[i] = rowmax;
  out_labels[i] = label;
}

// ---------------------------------------------------------------------------
// Launch: init colmax -> column-max pass -> assignment pass (all on stream).
// ---------------------------------------------------------------------------
extern "C" void kernel_launch(void* const* d_in, const int* in_sizes, int n_in,
                              void* d_out, int out_size, void* d_ws, size_t ws_size,
                              hipStream_t stream) {
  const float4* bboxes = (const float4*)d_in[0];
  const float* targets = (const float*)d_in[1];
  int N = in_sizes[0] / 4;
  int G = in_sizes[1] / 5;
  if (G > MAX_G) G = MAX_G;

  int* colmax = (int*)d_ws;
  float* out = (float*)d_out;

  hipLaunchKernelGGL(init_colmax_kernel, dim3(1), dim3(128), 0, stream, colmax, G);

  int nTiles = (N + 127) / 128;
  int g1 = nTiles < 1024 ? nTiles : 1024;
  if (g1 < 1) g1 = 1;
  hipLaunchKernelGGL(iou_colmax_kernel, dim3(g1), dim3(128), 0, stream,
                     bboxes, targets, colmax, N, G);

  int g2 = (N + 255) / 256;
  if (g2 < 1) g2 = 1;
  hipLaunchKernelGGL(assign_kernel, dim3(g2), dim3(256), 0, stream,
                     bboxes, targets, (const int*)colmax,
                     out, out + (size_t)N, out + 2 * (size_t)N, N, G);
}